// Attention_61701500174620
// MI455X (gfx1250) — compile-verified
//
#include <hip/hip_runtime.h>

typedef float v2f __attribute__((ext_vector_type(2)));
typedef float v8f __attribute__((ext_vector_type(8)));

#define LDS_STRIDE 68   // 64 + 4 pad; even (b64-aligned rows), conflict-free frag reads

// ---------------------------------------------------------------------------
// One 16x16 f32 output tile via V_WMMA_F32_16X16X4_F32, K=64 in 16 steps.
// A : 16xK row-major in LDS     (A[m*S + k])
// Bt: Kx16 stored TRANSPOSED    (Bt[n*S + k])  -> both operands are contiguous
//                                                 float2 (single ds_load_b64)
// ISA layouts (7.12.2):
//   A 16x4 : lane L holds row M=L%16, K = 2*(L/16) + vgpr
//   B 4x16 : lane L holds col N=L%16, K = vgpr + 2*(L/16)
//   C 16x16: lane L holds col N=L%16, row M = vgpr + 8*(L/16)
// ---------------------------------------------------------------------------
__device__ inline v8f gemm_tile_f32(const float* __restrict__ A,
                                    const float* __restrict__ Bt,
                                    int rb, int cb, int lane) {
  const int g = lane >> 4;       // lane group (0/1)
  const int n = lane & 15;       // row of A-tile / col of B-tile
  v8f acc = {};
#pragma unroll
  for (int kb = 0; kb < 64; kb += 4) {
    const v2f a = *(const v2f*)&A [(rb + n) * LDS_STRIDE + kb + 2 * g];
    const v2f b = *(const v2f*)&Bt[(cb + n) * LDS_STRIDE + kb + 2 * g];
    acc = __builtin_amdgcn_wmma_f32_16x16x4_f32(
        /*neg_a=*/false, a, /*neg_b=*/false, b,
        /*c_mod=*/(short)0, acc, /*reuse_a=*/false, /*reuse_b=*/false);
  }
  return acc;
}

// ---------------------------------------------------------------------------
// Kernel 1: fused attention per (b,c,d) slice.
//   S = softmax(Q @ K) ; y = S @ V + V   (all 64x64 fp32, LDS-resident)
// Writes y to d_out and accumulates per-channel sum / sum-of-squares.
// 128 threads = 4 wave32; wave w owns rows [16w, 16w+16).
// ---------------------------------------------------------------------------
__global__ __launch_bounds__(128) void attn_fused_kernel(
    const float* __restrict__ q, const float* __restrict__ k,
    const float* __restrict__ v, float* __restrict__ y,
    float* __restrict__ stats)   // stats[0..63]=sum, stats[64..127]=sumsq
{
  __shared__ float sQ [64 * LDS_STRIDE];   // Q, row-major  (A of GEMM1)
  __shared__ float sKt[64 * LDS_STRIDE];   // K^T: sKt[n*S + w]  (Bt of GEMM1)
  __shared__ float sVt[64 * LDS_STRIDE];   // V^T: sVt[n*S + h]  (Bt of GEMM2)
  __shared__ float sS [64 * LDS_STRIDE];   // scores, row-major  (A of GEMM2)
  __shared__ float redSum, redSq;

  const int tid   = threadIdx.x;
  const int lane  = tid & 31;
  const int wave  = tid >> 5;
  const int slice = blockIdx.x;                  // ((b*64 + c)*16 + d)
  const int chan  = (slice >> 4) & 63;           // channel c
  const size_t base = (size_t)slice * 4096;      // 64*64 elements per slice

  if (tid == 0) { redSum = 0.f; redSq = 0.f; }

  // ---- stage Q (row-major), K^T, V^T into padded LDS ----
#pragma unroll
  for (int i = 0; i < 8; ++i) {
    const int f   = tid + i * 128;               // float4 index, 0..1023
    const int row = f >> 4;                      // source row (K-dim index)
    const int c4  = f & 15;                      // source col/4
    const float4 qv = ((const float4*)(q + base))[f];
    const float4 kv = ((const float4*)(k + base))[f];
    const float4 vv = ((const float4*)(v + base))[f];
    *(float4*)&sQ[row * LDS_STRIDE + c4 * 4] = qv;
    const int c0 = c4 * 4;
    sKt[(c0 + 0) * LDS_STRIDE + row] = kv.x;
    sKt[(c0 + 1) * LDS_STRIDE + row] = kv.y;
    sKt[(c0 + 2) * LDS_STRIDE + row] = kv.z;
    sKt[(c0 + 3) * LDS_STRIDE + row] = kv.w;
    sVt[(c0 + 0) * LDS_STRIDE + row] = vv.x;
    sVt[(c0 + 1) * LDS_STRIDE + row] = vv.y;
    sVt[(c0 + 2) * LDS_STRIDE + row] = vv.z;
    sVt[(c0 + 3) * LDS_STRIDE + row] = vv.w;
  }
  __syncthreads();

  // ---- GEMM 1: S[16-row strip] = Q_strip @ K ----
  const int rb = wave * 16;
  const int g  = lane >> 4;
  const int n  = lane & 15;
#pragma unroll
  for (int j = 0; j < 4; ++j) {
    v8f acc = gemm_tile_f32(sQ, sKt, rb, j * 16, lane);
#pragma unroll
    for (int r = 0; r < 8; ++r)
      sS[(rb + r + 8 * g) * LDS_STRIDE + j * 16 + n] = acc[r];
  }
  __syncthreads();

  // ---- row softmax over S (thread t handles row t) ----
  if (tid < 64) {
    float* row = &sS[tid * LDS_STRIDE];
    float m = row[0];
#pragma unroll 4
    for (int i = 1; i < 64; ++i) m = fmaxf(m, row[i]);
    float s = 0.f;
#pragma unroll 4
    for (int i = 0; i < 64; ++i) { float e = __expf(row[i] - m); row[i] = e; s += e; }
    const float inv = 1.0f / s;
#pragma unroll 4
    for (int i = 0; i < 64; ++i) row[i] *= inv;
  }
  __syncthreads();

  // ---- GEMM 2: y_strip = S_strip @ V + V_strip ; write y, reduce stats ----
  float lsum = 0.f, lsq = 0.f;
#pragma unroll
  for (int j = 0; j < 4; ++j) {
    v8f acc = gemm_tile_f32(sS, sVt, rb, j * 16, lane);
#pragma unroll
    for (int r = 0; r < 8; ++r) {
      const int row = rb + r + 8 * g;
      const int col = j * 16 + n;
      const float yv = acc[r] + sVt[col * LDS_STRIDE + row];   // V[row][col]
      y[base + row * 64 + col] = yv;
      lsum += yv;
      lsq  += yv * yv;
    }
  }
  atomicAdd(&redSum, lsum);   // ds_add_f32
  atomicAdd(&redSq,  lsq);
  __syncthreads();
  if (tid == 0) {
    atomicAdd(&stats[chan],      redSum);   // global_atomic_add_f32
    atomicAdd(&stats[64 + chan], redSq);
  }
}

// ---------------------------------------------------------------------------
// Kernel 0: zero the stats accumulators (ws is poisoned, not re-zeroed).
// ---------------------------------------------------------------------------
__global__ void zero_stats_kernel(float* __restrict__ stats) {
  stats[threadIdx.x] = 0.f;   // 128 threads: sum + sumsq
}

// ---------------------------------------------------------------------------
// Kernel 2: fold mean/var into per-channel scale/shift.
//   scale = gamma * rsqrt(var + eps), shift = beta - mean*scale
// ---------------------------------------------------------------------------
__global__ void finalize_stats_kernel(float* __restrict__ stats,
                                      const float* __restrict__ gamma,
                                      const float* __restrict__ beta) {
  const int c = threadIdx.x;          // 64 threads
  const float N = 524288.f;           // B*D*H*W = 8*16*64*64
  const float mean = stats[c] / N;
  const float var  = stats[64 + c] / N - mean * mean;
  const float rs   = rsqrtf(var + 1e-5f);
  const float sc   = gamma[c] * rs;
  stats[128 + c] = sc;
  stats[192 + c] = beta[c] - mean * sc;
}

// ---------------------------------------------------------------------------
// Kernel 3: in-place BN + SiLU over d_out, float4 vectorized.
// channel of float4 index i is (i / 16384) % 64  (D*H*W = 65536 elems).
// ---------------------------------------------------------------------------
__global__ __launch_bounds__(256) void bn_silu_kernel(
    float* __restrict__ y, const float* __restrict__ stats) {
  const int idx = blockIdx.x * 256 + threadIdx.x;     // float4 index
  const int c   = (idx >> 14) & 63;
  const float sc = stats[128 + c];
  const float sh = stats[192 + c];
  float4 t = ((const float4*)y)[idx];
  float z;
  z = t.x * sc + sh; t.x = z / (1.f + __expf(-z));
  z = t.y * sc + sh; t.y = z / (1.f + __expf(-z));
  z = t.z * sc + sh; t.z = z / (1.f + __expf(-z));
  z = t.w * sc + sh; t.w = z / (1.f + __expf(-z));
  ((float4*)y)[idx] = t;
}

// ---------------------------------------------------------------------------
extern "C" void kernel_launch(void* const* d_in, const int* in_sizes, int n_in,
                              void* d_out, int out_size, void* d_ws, size_t ws_size,
                              hipStream_t stream) {
  const float* q     = (const float*)d_in[0];
  const float* k     = (const float*)d_in[1];
  const float* v     = (const float*)d_in[2];
  const float* gamma = (const float*)d_in[3];
  const float* beta  = (const float*)d_in[4];
  float* out   = (float*)d_out;
  float* stats = (float*)d_ws;          // 256 floats: sum | sumsq | scale | shift

  zero_stats_kernel<<<1, 128, 0, stream>>>(stats);
  attn_fused_kernel<<<8192, 128, 0, stream>>>(q, k, v, out, stats);
  finalize_stats_kernel<<<1, 64, 0, stream>>>(stats, gamma, beta);
  bn_silu_kernel<<<32768, 256, 0, stream>>>(out, stats);   // 33,554,432 / 4 / 256
}